// GAT_19911468384633
// MI455X (gfx1250) — compile-verified
//
#include <hip/hip_runtime.h>
#include <math.h>

#define N_NODES   50000
#define N_EDGES   800000
#define N_FEAT    128
#define N_HID     64
#define HEADS     4
#define HEAD_DIM  16
#define N_CLASS   10
#define NUM_GRAPHS 64
#define NEG_SLOPE 0.2f
#define KEY_NEG_INF 0x007FFFFFu   // order-key of -inf

typedef __attribute__((ext_vector_type(2))) float v2f;
typedef __attribute__((ext_vector_type(8))) float v8f;

__device__ __forceinline__ unsigned f_key(float f) {
  unsigned u = __float_as_uint(f);
  return (u & 0x80000000u) ? ~u : (u | 0x80000000u);
}
__device__ __forceinline__ float f_unkey(unsigned k) {
  return (k & 0x80000000u) ? __uint_as_float(k ^ 0x80000000u)
                           : __uint_as_float(~k);
}

// ---------------------------------------------------------------------------
// C[M,64] = A[M,K] @ B[K,64] with V_WMMA_F32_16X16X4_F32 (full fp32).
// One wave32 computes a 16x64 tile: 4 accumulators, K/4 wmma steps each.
// A frag: lanes 0-15 -> rows, K pair {k0,k0+1}; lanes 16-31 -> K pair {k0+2,k0+3}.
// B frag: vgpr0 = rows {k0, k0+2} striped over lane halves, vgpr1 = {k0+1, k0+3}.
// C/D: vgpr i -> M = half*8 + i, N = lane&15 (+16*tile).
// ---------------------------------------------------------------------------
__global__ void gemm64_wmma_f32(const float* __restrict__ A,
                                const float* __restrict__ B,
                                float* __restrict__ C, int M, int K)
{
  const int wave = blockIdx.x * (blockDim.x >> 5) + (threadIdx.x >> 5);
  const int row0 = wave << 4;
  if (row0 >= M) return;                 // whole-wave uniform exit: EXEC stays all-1s
  const int lane = threadIdx.x & 31;
  const int half = lane >> 4;
  const int lm   = lane & 15;

  v8f acc[4];
#pragma unroll
  for (int t = 0; t < 4; ++t) acc[t] = (v8f){0.f,0.f,0.f,0.f,0.f,0.f,0.f,0.f};

  const float* arow = A + (size_t)(row0 + lm) * K + 2 * half;
  for (int k0 = 0; k0 < K; k0 += 4) {
    v2f a;
    a.x = arow[k0];
    a.y = arow[k0 + 1];
    const float* brow0 = B + (size_t)(k0 + 2 * half) * N_HID;
    const float* brow1 = brow0 + N_HID;
#pragma unroll
    for (int t = 0; t < 4; ++t) {
      v2f b;
      b.x = brow0[t * 16 + lm];
      b.y = brow1[t * 16 + lm];
      acc[t] = __builtin_amdgcn_wmma_f32_16x16x4_f32(
          false, a, false, b, (short)0, acc[t], false, false);
    }
  }

#pragma unroll
  for (int t = 0; t < 4; ++t) {
    const int n = t * 16 + lm;
#pragma unroll
    for (int i = 0; i < 8; ++i)
      C[(size_t)(row0 + half * 8 + i) * N_HID + n] = acc[t][i];
  }
}

// ---------------------------------------------------------------------------
// Per-node attention logits: alpha_src/dst[n,h] = <h[n, h*16 : h*16+16], a[h]>
// ---------------------------------------------------------------------------
__global__ void node_alpha(const float* __restrict__ h,
                           const float* __restrict__ a_src,
                           const float* __restrict__ a_dst,
                           float* __restrict__ out_s,
                           float* __restrict__ out_d)
{
  int t = blockIdx.x * blockDim.x + threadIdx.x;
  if (t >= N_NODES * HEADS) return;
  int n = t >> 2, hh = t & 3;
  const float* hp = h + (size_t)n * N_HID + hh * HEAD_DIM;
  const float* ap = a_src + hh * HEAD_DIM;
  const float* bp = a_dst + hh * HEAD_DIM;
  float s = 0.f, d = 0.f;
#pragma unroll
  for (int c = 0; c < HEAD_DIM; ++c) {
    float v = hp[c];
    s += v * ap[c];
    d += v * bp[c];
  }
  out_s[t] = s;
  out_d[t] = d;
}

__global__ void init_seg(unsigned* __restrict__ mkey, float* __restrict__ denom)
{
  int t = blockIdx.x * blockDim.x + threadIdx.x;
  if (t >= N_NODES * HEADS) return;
  mkey[t] = KEY_NEG_INF;
  denom[t] = 0.f;
}

// Pass A: edge logits + leaky-relu, segment-max (monotone uint key atomicMax)
__global__ void edge_logit_max(const int* __restrict__ src,
                               const int* __restrict__ dst,
                               const float* __restrict__ as_,
                               const float* __restrict__ ad_,
                               float* __restrict__ ebuf,
                               unsigned* __restrict__ mkey)
{
  int t = blockIdx.x * blockDim.x + threadIdx.x;
  if (t >= N_EDGES * HEADS) return;
  int e = t >> 2, hh = t & 3;
  int s = src[e], d = dst[e];
  float v = as_[s * HEADS + hh] + ad_[d * HEADS + hh];
  v = (v > 0.f) ? v : NEG_SLOPE * v;
  ebuf[t] = v;
  atomicMax(&mkey[d * HEADS + hh], f_key(v));
}

// Pass B: w = exp(e - m[dst]); segment-sum denominator
__global__ void edge_exp_sum(const int* __restrict__ dst,
                             const unsigned* __restrict__ mkey,
                             float* __restrict__ ebuf,
                             float* __restrict__ denom)
{
  int t = blockIdx.x * blockDim.x + threadIdx.x;
  if (t >= N_EDGES * HEADS) return;
  int e = t >> 2, hh = t & 3;
  int d = dst[e];
  float m = f_unkey(mkey[d * HEADS + hh]);
  float w = expf(ebuf[t] - m);
  ebuf[t] = w;
  atomicAdd(&denom[d * HEADS + hh], w);
}

// Pass C: out[dst, c] += h[src, c] * (w / (denom[dst] + 1e-16)), 64 lanes/edge
__global__ void edge_aggregate(const int* __restrict__ src,
                               const int* __restrict__ dst,
                               const float* __restrict__ hbuf,
                               const float* __restrict__ ebuf,
                               const float* __restrict__ denom,
                               float* __restrict__ outb)
{
  long long t = (long long)blockIdx.x * blockDim.x + threadIdx.x;
  if (t >= (long long)N_EDGES * N_HID) return;
  int e = (int)(t >> 6);
  int c = (int)(t & 63);
  int hh = c >> 4;
  int s = src[e], d = dst[e];
  float w   = ebuf[e * HEADS + hh];
  float den = denom[d * HEADS + hh];
  float alpha = w / (den + 1e-16f);
  atomicAdd(&outb[(size_t)d * N_HID + c], hbuf[(size_t)s * N_HID + c] * alpha);
}

__global__ void bias_relu(const float* __restrict__ agg,
                          const float* __restrict__ bias,
                          float* __restrict__ x)
{
  int t = blockIdx.x * blockDim.x + threadIdx.x;
  if (t >= N_NODES * N_HID) return;
  float v = agg[t] + bias[t & 63];
  x[t] = v > 0.f ? v : 0.f;
}

__global__ void pool_scatter(const float* __restrict__ x,
                             const int* __restrict__ batch,
                             float* __restrict__ sums,
                             float* __restrict__ cnt)
{
  int t = blockIdx.x * blockDim.x + threadIdx.x;
  if (t >= N_NODES * N_HID) return;
  int n = t >> 6, c = t & 63;
  int g = batch[n];
  atomicAdd(&sums[g * N_HID + c], x[t]);
  if (c == 0) atomicAdd(&cnt[g], 1.f);
}

__global__ void classify(const float* __restrict__ sums,
                         const float* __restrict__ cnt,
                         const float* __restrict__ fcW,
                         const float* __restrict__ fcb,
                         float* __restrict__ out)
{
  int t = blockIdx.x * blockDim.x + threadIdx.x;
  if (t >= NUM_GRAPHS * N_CLASS) return;
  int g = t / N_CLASS, k = t % N_CLASS;
  float inv = 1.f / fmaxf(cnt[g], 1.f);
  float acc = fcb[k];
#pragma unroll
  for (int c = 0; c < N_HID; ++c)
    acc += (sums[g * N_HID + c] * inv) * fcW[c * N_CLASS + k];
  out[t] = acc;
}

// ---------------------------------------------------------------------------
static void run_gat_layer(const float* xin, int K, const float* W,
                          const float* a_s, const float* a_d, const float* bias,
                          float* bufH, float* bufA, float* bufX,
                          float* as_, float* ad_, float* ebuf,
                          unsigned* mkey, float* denom,
                          const int* src, const int* dst, hipStream_t stream)
{
  const int waves  = N_NODES / 16;                 // 3125, exact
  const int gblk   = (waves + 7) / 8;              // 8 waves / 256-thread block
  gemm64_wmma_f32<<<gblk, 256, 0, stream>>>(xin, W, bufH, N_NODES, K);

  node_alpha<<<(N_NODES * HEADS + 255) / 256, 256, 0, stream>>>(bufH, a_s, a_d, as_, ad_);

  hipMemsetAsync(bufA, 0, (size_t)N_NODES * N_HID * sizeof(float), stream);
  init_seg<<<(N_NODES * HEADS + 255) / 256, 256, 0, stream>>>(mkey, denom);

  edge_logit_max<<<(N_EDGES * HEADS + 255) / 256, 256, 0, stream>>>(src, dst, as_, ad_, ebuf, mkey);
  edge_exp_sum  <<<(N_EDGES * HEADS + 255) / 256, 256, 0, stream>>>(dst, mkey, ebuf, denom);

  long long aggT = (long long)N_EDGES * N_HID;     // 51.2M threads
  edge_aggregate<<<(unsigned)((aggT + 255) / 256), 256, 0, stream>>>(src, dst, bufH, ebuf, denom, bufA);

  bias_relu<<<(N_NODES * N_HID + 255) / 256, 256, 0, stream>>>(bufA, bias, bufX);
}

extern "C" void kernel_launch(void* const* d_in, const int* in_sizes, int n_in,
                              void* d_out, int out_size, void* d_ws, size_t ws_size,
                              hipStream_t stream)
{
  const float* feat = (const float*)d_in[0];
  const float* W1   = (const float*)d_in[1];
  const float* a1s  = (const float*)d_in[2];
  const float* a1d  = (const float*)d_in[3];
  const float* b1   = (const float*)d_in[4];
  const float* W2   = (const float*)d_in[5];
  const float* a2s  = (const float*)d_in[6];
  const float* a2d  = (const float*)d_in[7];
  const float* b2   = (const float*)d_in[8];
  const float* fcW  = (const float*)d_in[9];
  const float* fcb  = (const float*)d_in[10];
  const int*   eidx = (const int*)d_in[11];
  const int*   batch= (const int*)d_in[12];
  const int* src = eidx;
  const int* dst = eidx + N_EDGES;

  const size_t NH = (size_t)N_NODES * N_HID;
  float*    bufH  = (float*)d_ws;
  float*    bufA  = bufH + NH;
  float*    bufX  = bufA + NH;
  float*    as_   = bufX + NH;
  float*    ad_   = as_  + (size_t)N_NODES * HEADS;
  float*    ebuf  = ad_  + (size_t)N_NODES * HEADS;
  unsigned* mkey  = (unsigned*)(ebuf + (size_t)N_EDGES * HEADS);
  float*    denom = (float*)(mkey + (size_t)N_NODES * HEADS);
  float*    sums  = denom + (size_t)N_NODES * HEADS;
  float*    cnt   = sums + (size_t)NUM_GRAPHS * N_HID;

  // Layer 1: feat[50000,128] -> x1
  run_gat_layer(feat, N_FEAT, W1, a1s, a1d, b1,
                bufH, bufA, bufX, as_, ad_, ebuf, mkey, denom, src, dst, stream);
  // Layer 2: x1[50000,64] -> x2 (bufX overwritten in place after GEMM consumed it)
  run_gat_layer(bufX, N_HID, W2, a2s, a2d, b2,
                bufH, bufA, bufX, as_, ad_, ebuf, mkey, denom, src, dst, stream);

  // Global mean pool + classifier
  hipMemsetAsync(sums, 0, (size_t)(NUM_GRAPHS * N_HID + NUM_GRAPHS) * sizeof(float), stream);
  pool_scatter<<<(N_NODES * N_HID + 255) / 256, 256, 0, stream>>>(bufX, batch, sums, cnt);
  classify<<<(NUM_GRAPHS * N_CLASS + 255) / 256, 256, 0, stream>>>(sums, cnt, fcW, fcb, (float*)d_out);
}